// ATTN_32461362823930
// MI455X (gfx1250) — compile-verified
//
#include <hip/hip_runtime.h>
#include <hip/hip_bf16.h>
#include <math.h>

typedef float v2f __attribute__((ext_vector_type(2)));
typedef float v4f __attribute__((ext_vector_type(4)));
typedef float v8f __attribute__((ext_vector_type(8)));

#define SEQ   2048
#define BATCH 32
#define DIM   1024
#define NCHUNK 64
#define CH    (SEQ / NCHUNK)   // 32 rows per chunk
#define RB    4                // rows per reduction batch

// ---------------------------------------------------------------------------
// Kernel 1: semi[b,d] = sum_k h[b,k] * Wi[d,k]   (M=32, N=1024, K=1024)
// One wave per 16x16 tile, V_WMMA_F32_16X16X4_F32, K stepped by 4.
// A frag: lane holds A[lane%16][k0 + 2*(lane/16) + {0,1}]  (float2)
// B frag: lane holds B[k0 + 2*(lane/16) + {0,1}][lane%16] = Wi[n][k] (float2)
// D frag: VGPR r -> row (r + 8*(lane/16)), col lane%16
// ---------------------------------------------------------------------------
__global__ void attn_gemm_semi(const float* __restrict__ h,
                               const float* __restrict__ Wi,
                               float* __restrict__ semi) {
  const int lane = threadIdx.x & 31;
  const int wv   = threadIdx.x >> 5;                 // 0..3
  const int n0   = (blockIdx.x * 4 + wv) * 16;
  const int m0   = blockIdx.y * 16;
  const int mn   = lane & 15;
  const int koff = (lane >> 4) << 1;                 // 0 or 2

  const float* __restrict__ arow = h  + (size_t)(m0 + mn) * DIM;
  const float* __restrict__ brow = Wi + (size_t)(n0 + mn) * DIM;

  v8f c = {};
  for (int k0 = 0; k0 < DIM; k0 += 4) {
    v2f a = *(const v2f*)(arow + k0 + koff);
    v2f b = *(const v2f*)(brow + k0 + koff);
    c = __builtin_amdgcn_wmma_f32_16x16x4_f32(false, a, false, b,
                                              (short)0, c, false, false);
  }
  const int row = m0 + ((lane >> 4) << 3);
  const int col = n0 + mn;
#pragma unroll
  for (int r = 0; r < 8; ++r)
    semi[(size_t)(row + r) * DIM + col] = c[r];
}

// ---------------------------------------------------------------------------
// Kernel 2: single streaming pass over input with online softmax.
// Block = (b, chunk). 256 threads; thread t owns d = 4t..4t+3.
// For each row s in chunk: logit = <input[s,b,:], semi[b,:]> (block reduce),
// then rescale-accumulate exp-weighted row kept in registers.
// Emits per-chunk partials: acc[D], running max m, running sum Z.
// ---------------------------------------------------------------------------
__global__ void attn_flash(const float* __restrict__ input,
                           const float* __restrict__ semi,
                           float* __restrict__ pacc,
                           float* __restrict__ pmz) {
  const int b     = blockIdx.y;
  const int chunk = blockIdx.x;
  const int t     = threadIdx.x;
  const int lane  = t & 31;
  const int wid   = t >> 5;

  __shared__ float red[RB][8];
  __shared__ float lbc[RB];

  const v4f sm = ((const v4f*)(semi + (size_t)b * DIM))[t];

  float m = -INFINITY, Z = 0.f;
  v4f acc = {0.f, 0.f, 0.f, 0.f};
  const int s0 = chunk * CH;

  for (int r0 = 0; r0 < CH; r0 += RB) {
    v4f   x[RB];
    float loc[RB];
#pragma unroll
    for (int r = 0; r < RB; ++r) {
      const int s = s0 + r0 + r;
      const v4f* __restrict__ rowp =
          (const v4f*)(input + ((size_t)s * BATCH + b) * DIM);
      x[r] = __builtin_nontemporal_load(rowp + t);   // streamed, read-once
      loc[r] = x[r].x * sm.x + x[r].y * sm.y + x[r].z * sm.z + x[r].w * sm.w;
    }
    // wave32 reduction
#pragma unroll
    for (int r = 0; r < RB; ++r)
#pragma unroll
      for (int off = 16; off > 0; off >>= 1)
        loc[r] += __shfl_xor(loc[r], off, 32);
    if (lane == 0) {
#pragma unroll
      for (int r = 0; r < RB; ++r) red[r][wid] = loc[r];
    }
    __syncthreads();
    if (t < RB) {
      float s = 0.f;
#pragma unroll
      for (int w = 0; w < 8; ++w) s += red[t][w];
      lbc[t] = s;
    }
    __syncthreads();

    float l[RB];
    float lmax = -INFINITY;
#pragma unroll
    for (int r = 0; r < RB; ++r) { l[r] = lbc[r]; lmax = fmaxf(lmax, l[r]); }
    const float mn = fmaxf(m, lmax);
    const float sc = __expf(m - mn);                 // exp(-inf)=0 on first pass
    float wx = 0.f, wy = 0.f, wz = 0.f, ww = 0.f, zs = 0.f;
#pragma unroll
    for (int r = 0; r < RB; ++r) {
      const float w = __expf(l[r] - mn);
      zs += w;
      wx += w * x[r].x; wy += w * x[r].y; wz += w * x[r].z; ww += w * x[r].w;
    }
    Z = Z * sc + zs;
    acc.x = acc.x * sc + wx;
    acc.y = acc.y * sc + wy;
    acc.z = acc.z * sc + wz;
    acc.w = acc.w * sc + ww;
    m = mn;
    __syncthreads();   // red/lbc reused next batch
  }

  float* __restrict__ pa = pacc + ((size_t)b * NCHUNK + chunk) * DIM;
  ((v4f*)pa)[t] = acc;
  if (t == 0) {
    pmz[((size_t)b * NCHUNK + chunk) * 2 + 0] = m;
    pmz[((size_t)b * NCHUNK + chunk) * 2 + 1] = Z;
  }
}

// ---------------------------------------------------------------------------
// Kernel 3: merge per-chunk partials into s_tilde[b,:].
// One block per b; thread t owns d = 4t..4t+3.
// ---------------------------------------------------------------------------
__global__ void attn_combine(const float* __restrict__ pacc,
                             const float* __restrict__ pmz,
                             float* __restrict__ stilde) {
  const int b = blockIdx.x;
  const int t = threadIdx.x;

  float m = -INFINITY;
  for (int c = 0; c < NCHUNK; ++c)
    m = fmaxf(m, pmz[((size_t)b * NCHUNK + c) * 2 + 0]);

  float Z = 0.f;
  for (int c = 0; c < NCHUNK; ++c)
    Z += pmz[((size_t)b * NCHUNK + c) * 2 + 1] *
         __expf(pmz[((size_t)b * NCHUNK + c) * 2 + 0] - m);

  float sx = 0.f, sy = 0.f, sz = 0.f, sw = 0.f;
  for (int c = 0; c < NCHUNK; ++c) {
    const float w = __expf(pmz[((size_t)b * NCHUNK + c) * 2 + 0] - m);
    const v4f a = ((const v4f*)(pacc + ((size_t)b * NCHUNK + c) * DIM))[t];
    sx += w * a.x; sy += w * a.y; sz += w * a.z; sw += w * a.w;
  }
  const float inv = 1.f / Z;
  v4f o;
  o.x = sx * inv; o.y = sy * inv; o.z = sz * inv; o.w = sw * inv;
  ((v4f*)(stilde + (size_t)b * DIM))[t] = o;
}

// ---------------------------------------------------------------------------
// Kernel 4: c[b,n] = tanh( sum_k [s_tilde|h][b,k] * Wo[n,k] )
// M=32, N=1024, K=2048; split K loop at 1024 to switch A source.
// ---------------------------------------------------------------------------
__global__ void attn_gemm_out(const float* __restrict__ stilde,
                              const float* __restrict__ h,
                              const float* __restrict__ Wo,
                              float* __restrict__ out) {
  const int lane = threadIdx.x & 31;
  const int wv   = threadIdx.x >> 5;
  const int n0   = (blockIdx.x * 4 + wv) * 16;
  const int m0   = blockIdx.y * 16;
  const int mn   = lane & 15;
  const int koff = (lane >> 4) << 1;

  const float* __restrict__ a0   = stilde + (size_t)(m0 + mn) * DIM;
  const float* __restrict__ a1   = h      + (size_t)(m0 + mn) * DIM;
  const float* __restrict__ brow = Wo     + (size_t)(n0 + mn) * (2 * DIM);

  v8f c = {};
  for (int k0 = 0; k0 < DIM; k0 += 4) {
    v2f a = *(const v2f*)(a0 + k0 + koff);
    v2f b = *(const v2f*)(brow + k0 + koff);
    c = __builtin_amdgcn_wmma_f32_16x16x4_f32(false, a, false, b,
                                              (short)0, c, false, false);
  }
  for (int k0 = 0; k0 < DIM; k0 += 4) {
    v2f a = *(const v2f*)(a1 + k0 + koff);
    v2f b = *(const v2f*)(brow + DIM + k0 + koff);
    c = __builtin_amdgcn_wmma_f32_16x16x4_f32(false, a, false, b,
                                              (short)0, c, false, false);
  }
  const int row = m0 + ((lane >> 4) << 3);
  const int col = n0 + mn;
#pragma unroll
  for (int r = 0; r < 8; ++r)
    out[(size_t)(row + r) * DIM + col] = tanhf(c[r]);
}

// ---------------------------------------------------------------------------
extern "C" void kernel_launch(void* const* d_in, const int* in_sizes, int n_in,
                              void* d_out, int out_size, void* d_ws, size_t ws_size,
                              hipStream_t stream) {
  const float* input = (const float*)d_in[0];   // [S, B, D]
  const float* h     = (const float*)d_in[1];   // [B, D]
  const float* Wi    = (const float*)d_in[2];   // [D, D]
  const float* Wo    = (const float*)d_in[3];   // [D, 2D]
  float* out = (float*)d_out;                   // [B, D]

  float* ws     = (float*)d_ws;
  float* semi   = ws;                                   // 32*1024
  float* stilde = semi + (size_t)BATCH * DIM;           // 32*1024
  float* pmz    = stilde + (size_t)BATCH * DIM;         // 32*64*2
  float* pacc   = pmz + (size_t)BATCH * NCHUNK * 2;     // 32*64*1024 (8 MB)

  attn_gemm_semi<<<dim3(16, 2), 128, 0, stream>>>(h, Wi, semi);
  attn_flash<<<dim3(NCHUNK, BATCH), 256, 0, stream>>>(input, semi, pacc, pmz);
  attn_combine<<<BATCH, 256, 0, stream>>>(pacc, pmz, stilde);
  attn_gemm_out<<<dim3(16, 2), 128, 0, stream>>>(stilde, h, Wo, out);
}